// DPTLiftSplatShoot_3169685865333
// MI455X (gfx1250) — compile-verified
//
#include <hip/hip_runtime.h>

// ---------------- problem constants (match reference) ----------------
#define BN    4
#define NN    6
#define DN    41
#define FH_   16
#define FW_   44
#define CN    64
#define PPC   (DN * FH_ * FW_)        // 28864 points per camera
#define NCAM  (BN * NN)               // 24
#define BLK   256
#define BPC   ((PPC + BLK - 1) / BLK) // 113 blocks per camera
#define NXY   256
#define SPAT  (NXY * NXY)             // 65536 voxels per batch
#define OUTN  (BN * CN * SPAT)        // 16,777,216 output floats

typedef float v2f __attribute__((ext_vector_type(2)));
typedef float v8f __attribute__((ext_vector_type(8)));

// 3x3 inverse via adjugate (inputs are well-conditioned rotations / intrinsics)
__device__ __forceinline__ void inv3(const float m[9], float o[9]) {
    float a = m[0], b = m[1], c = m[2];
    float d = m[3], e = m[4], f = m[5];
    float g = m[6], h = m[7], i = m[8];
    float A =  (e * i - f * h);
    float Bc = -(d * i - f * g);
    float Cc =  (d * h - e * g);
    float det = a * A + b * Bc + c * Cc;
    float r = 1.0f / det;
    o[0] = A * r;              o[1] = (c * h - b * i) * r; o[2] = (b * f - c * e) * r;
    o[3] = Bc * r;             o[4] = (a * i - c * g) * r; o[5] = (c * d - a * f) * r;
    o[6] = Cc * r;             o[7] = (b * g - a * h) * r; o[8] = (a * e - b * d) * r;
}

// ---------------- zero-fill accumulator ----------------
__global__ void __launch_bounds__(BLK) lss_zero(float4* __restrict__ p, int n4) {
    int i = blockIdx.x * BLK + threadIdx.x;
    if (i < n4) p[i] = make_float4(0.f, 0.f, 0.f, 0.f);
}

// ---------------- main: geometry (WMMA f32) + merged wave-cooperative scatter ----------------
__global__ void __launch_bounds__(BLK) lss_main(
    const float* __restrict__ x,          // (B,N,D,FH,FW,C)
    const float* __restrict__ rots,       // (B,N,3,3)
    const float* __restrict__ trans,      // (B,N,3)
    const float* __restrict__ intrins,    // (B,N,3,3)
    const float* __restrict__ post_rots,  // (B,N,3,3)
    const float* __restrict__ post_trans, // (B,N,3)
    const float* __restrict__ frustum,    // (D,FH,FW,3)
    float* __restrict__ dst,              // staged: [b][y][x][c]  direct: [b][c][y][x]
    int staged)
{
    const int cam  = blockIdx.x / BPC;    // uniform per block
    const int blk  = blockIdx.x % BPC;
    const int tid  = threadIdx.x;
    const int lane = tid & 31;
    const int b    = cam / NN;

    int pid = blk * BLK + tid;
    const bool valid = (pid < PPC);
    if (!valid) pid = PPC - 1;            // clamp: keep EXEC all-ones for WMMA

    // ---- per-camera matrices (block-uniform) ----
    float K[9], PR[9], R[9], Ki[9], PRi[9], comb[9], T[3], PT[3];
#pragma unroll
    for (int i = 0; i < 9; ++i) {
        K[i]  = intrins[cam * 9 + i];
        PR[i] = post_rots[cam * 9 + i];
        R[i]  = rots[cam * 9 + i];
    }
#pragma unroll
    for (int i = 0; i < 3; ++i) {
        T[i]  = trans[cam * 3 + i];
        PT[i] = post_trans[cam * 3 + i];
    }
    inv3(K, Ki);
    inv3(PR, PRi);
#pragma unroll
    for (int i = 0; i < 3; ++i)
#pragma unroll
        for (int j = 0; j < 3; ++j)
            comb[i * 3 + j] = R[i * 3 + 0] * Ki[0 * 3 + j]
                            + R[i * 3 + 1] * Ki[1 * 3 + j]
                            + R[i * 3 + 2] * Ki[2 * 3 + j];

    // ---- pre-transform point q = (u*z, v*z, z) after post-roto/trans undo ----
    const float* F = frustum + (size_t)pid * 3; // pid == ((d*FH + h)*FW + w)
    float p0x = F[0] - PT[0], p0y = F[1] - PT[1], p0z = F[2] - PT[2];
    float p1x = PRi[0] * p0x + PRi[1] * p0y + PRi[2] * p0z;
    float p1y = PRi[3] * p0x + PRi[4] * p0y + PRi[5] * p0z;
    float p1z = PRi[6] * p0x + PRi[7] * p0y + PRi[8] * p0z;
    float qx = p1x * p1z, qy = p1y * p1z, qz = p1z;

    // ---- P = [comb | T] * (qx,qy,qz,1)  via V_WMMA_F32_16X16X4_F32 ----
    // A (16x4, MxK): lanes 0-15 hold M=lane {V0=K0,V1=K1}; lanes 16-31 {V0=K2,V1=K3}
    v2f a; a[0] = 0.f; a[1] = 0.f;
    {
        int m = lane & 15;
        if (m < 3) {
            if (lane < 16) { a[0] = comb[m * 3 + 0]; a[1] = comb[m * 3 + 1]; }
            else           { a[0] = comb[m * 3 + 2]; a[1] = T[m]; }
        }
    }
    // B (4x16, KxN): lanes 0-15 hold column N=lane {V0=K0,V1=K1}; lanes 16-31 {V0=K2,V1=K3}
    float oqx = __shfl_xor(qx, 16);
    float oqy = __shfl_xor(qy, 16);
    float oqz = __shfl_xor(qz, 16);
    v2f b0, b1;
    if (lane < 16) { b0[0] = qx;  b0[1] = qy;   b1[0] = oqx; b1[1] = oqy; }
    else           { b0[0] = oqz; b0[1] = 1.0f; b1[0] = qz;  b1[1] = 1.0f; }
    v8f cz = {0.f, 0.f, 0.f, 0.f, 0.f, 0.f, 0.f, 0.f};
    v8f d0 = __builtin_amdgcn_wmma_f32_16x16x4_f32(false, a, false, b0, (short)0, cz, false, false);
    v8f d1 = __builtin_amdgcn_wmma_f32_16x16x4_f32(false, a, false, b1, (short)0, cz, false, false);
    // D (16x16): lane n<16 holds D[M=j][N=n] in v[j]; group-1 results shuffled back up
    float sx = __shfl_xor(d1[0], 16);
    float sy = __shfl_xor(d1[1], 16);
    float sz = __shfl_xor(d1[2], 16);
    float Px = (lane < 16) ? d0[0] : sx;
    float Py = (lane < 16) ? d0[1] : sy;
    float Pz = (lane < 16) ? d0[2] : sz;

    // ---- voxelize: trunc((P - (BX - DX/2)) / DX), exact f32 arithmetic ----
    const float lx = -51.0f - 0.2f;
    const float ly = -51.0f - 0.2f;
    const float lz =   0.0f - 10.0f;
    int g0 = (int)((Px - lx) / 0.4f);
    int g1 = (int)((Py - ly) / 0.4f);
    int g2 = (int)((Pz - lz) / 20.0f);
    bool kept = valid && (g0 >= 0) && (g0 < NXY) && (g1 >= 0) && (g1 < NXY) && (g2 == 0);

    unsigned vbase = 0xFFFFFFFFu;
    if (kept)
        vbase = staged ? (unsigned)((b * SPAT + g1 * NXY + g0) * CN)
                       : (unsigned)(b * (CN * SPAT) + g1 * NXY + g0);
    const size_t cstride = staged ? (size_t)1 : (size_t)SPAT;
    const int rowB = (cam * PPC + pid) * CN;   // feature row base (elements)

    // ---- wave-cooperative scatter with register merging of same-voxel runs ----
    const int c0 = lane * 2;                   // each lane owns channels c0, c0+1
    float a0 = 0.f, a1 = 0.f;
    unsigned cur = 0xFFFFFFFFu;
    for (int i = 0; i < 32; ++i) {
        unsigned vb = (unsigned)__builtin_amdgcn_readlane((int)vbase, i);
        int      rb = __builtin_amdgcn_readlane(rowB, i);
        if (vb == 0xFFFFFFFFu) continue;       // wave-uniform branch
        if (vb != cur) {
            if (cur != 0xFFFFFFFFu) {
                atomicAdd(dst + (size_t)cur + (size_t)c0 * cstride, a0);
                atomicAdd(dst + (size_t)cur + (size_t)(c0 + 1) * cstride, a1);
            }
            cur = vb; a0 = 0.f; a1 = 0.f;
        }
        const float2 v = ((const float2*)(x + (size_t)rb))[lane]; // coalesced 256B row
        a0 += v.x; a1 += v.y;
    }
    if (cur != 0xFFFFFFFFu) {
        atomicAdd(dst + (size_t)cur + (size_t)c0 * cstride, a0);
        atomicAdd(dst + (size_t)cur + (size_t)(c0 + 1) * cstride, a1);
    }
}

// ---------------- transpose ws[b][s][c] -> out[b][c][s]  (s = y*256+x) ----------------
__global__ void __launch_bounds__(BLK) lss_transpose(const float* __restrict__ ws,
                                                     float* __restrict__ out) {
    __shared__ float t[64][65];
    const int blk = blockIdx.x;
    const int b   = blk >> 10;            // 1024 tiles of 64 spatial rows per batch
    const int s0  = (blk & 1023) << 6;
    const float* src = ws + ((size_t)b * SPAT + s0) * CN;
#pragma unroll
    for (int k = 0; k < 16; ++k) {
        int idx = k * BLK + threadIdx.x;
        int sl = idx >> 6, c = idx & 63;
        t[sl][c] = src[sl * CN + c];      // coalesced over c
    }
    __syncthreads();
    float* dstb = out + (size_t)b * CN * SPAT + s0;
#pragma unroll
    for (int k = 0; k < 16; ++k) {
        int idx = k * BLK + threadIdx.x;
        int c = idx >> 6, sl = idx & 63;
        dstb[(size_t)c * SPAT + sl] = t[sl][c]; // coalesced over sl
    }
}

extern "C" void kernel_launch(void* const* d_in, const int* in_sizes, int n_in,
                              void* d_out, int out_size, void* d_ws, size_t ws_size,
                              hipStream_t stream) {
    const float* x          = (const float*)d_in[0];
    const float* rots       = (const float*)d_in[1];
    const float* trans      = (const float*)d_in[2];
    const float* intrins    = (const float*)d_in[3];
    const float* post_rots  = (const float*)d_in[4];
    const float* post_trans = (const float*)d_in[5];
    const float* frustum    = (const float*)d_in[6];
    float* out = (float*)d_out;

    const size_t need = (size_t)OUTN * sizeof(float);   // 64 MB accumulator
    const int staged = (ws_size >= need) ? 1 : 0;
    float* acc = staged ? (float*)d_ws : out;

    lss_zero<<<OUTN / 4 / BLK, BLK, 0, stream>>>((float4*)acc, OUTN / 4);
    lss_main<<<NCAM * BPC, BLK, 0, stream>>>(x, rots, trans, intrins, post_rots,
                                             post_trans, frustum, acc, staged);
    if (staged)
        lss_transpose<<<BN * (SPAT / 64), BLK, 0, stream>>>(acc, out);
}